// ChronoHybridLadderV2C_62801011802692
// MI455X (gfx1250) — compile-verified
//
#include <hip/hip_runtime.h>
#include <math.h>

typedef float v8f __attribute__((ext_vector_type(8)));
typedef float v2f __attribute__((ext_vector_type(2)));

namespace {
constexpr int Bc = 32, Sc = 2048, Hc = 1024;
constexpr int NCH = 8, SCH = Sc / NCH;   // S reduction chunks
constexpr float TEMP = 0.25f;
}

// ---------------- pooling over S ----------------
__global__ void pool_partial(const float* __restrict__ hidden, const int* __restrict__ mask,
                             float* __restrict__ partial) {
  int h = blockIdx.x * 256 + threadIdx.x;
  int chunk = blockIdx.y, b = blockIdx.z;
  const float* hp = hidden + ((size_t)b * Sc + (size_t)chunk * SCH) * Hc + h;
  const int* mp = mask + b * Sc + chunk * SCH;
  float acc = 0.f;
  for (int s = 0; s < SCH; ++s) acc += hp[(size_t)s * Hc] * (float)mp[s];
  partial[((size_t)b * NCH + chunk) * Hc + h] = acc;
}

__global__ void mask_count(const int* __restrict__ mask, float* __restrict__ cnt) {
  int b = blockIdx.x, tid = threadIdx.x;
  __shared__ float red[256];
  float s = 0.f;
  for (int i = tid; i < Sc; i += 256) s += (float)mask[b * Sc + i];
  red[tid] = s; __syncthreads();
  for (int o = 128; o > 0; o >>= 1) { if (tid < o) red[tid] += red[tid + o]; __syncthreads(); }
  if (tid == 0) cnt[b] = red[0];
}

__global__ void pool_finalize(const float* __restrict__ partial, const float* __restrict__ cnt,
                              const float* __restrict__ hidden, float* __restrict__ pooledLast) {
  int h = blockIdx.x * 256 + threadIdx.x; int b = blockIdx.y;
  float acc = 0.f;
  for (int c = 0; c < NCH; ++c) acc += partial[((size_t)b * NCH + c) * Hc + h];
  pooledLast[(size_t)b * 2 * Hc + h] = acc / fmaxf(cnt[b], 1.f);
  pooledLast[(size_t)b * 2 * Hc + Hc + h] = hidden[((size_t)b * Sc + (Sc - 1)) * Hc + h];
}

// ---------------- generic f32 WMMA GEMM: C[M,N] = act(A[M,K] @ W[K,N] + bias) -----------
// one wave per 16x16 tile. Branch-free main loop over K&~3; single guarded tail
// iteration only for K not divisible by 4 (the gate-feature GEMMs).
__global__ void gemm16_wmma(const float* __restrict__ A, const float* __restrict__ W,
                            const float* __restrict__ bias, float* __restrict__ C,
                            int M, int N, int K, int act) {
  int n0 = blockIdx.x * 16;
  int m0 = blockIdx.y * 16;
  int lane = threadIdx.x;
  int half = lane >> 4;                 // 0: K=k..k+1, 1: K=k+2..k+3
  int nc = n0 + (lane & 15);
  const float* arow = A + (size_t)(m0 + (lane & 15)) * K;
  v8f acc = {};

  int Kmain = K & ~3;
  const float* ap = arow + half * 2;
  const float* bp = W + (size_t)(half * 2) * N + nc;
  const size_t stepB = (size_t)4 * N;
  for (int k = 0; k < Kmain; k += 4) {
    v2f a, b;
    a.x = ap[0];
    a.y = ap[1];
    b.x = bp[0];
    b.y = bp[N];
    ap += 4;
    bp += stepB;
    acc = __builtin_amdgcn_wmma_f32_16x16x4_f32(false, a, false, b, (short)0, acc, false, false);
  }
  if (Kmain < K) {                      // K tail (1..3 leftover), guarded loads
    int ka = Kmain + half * 2;
    v2f a, b;
    a.x = (ka     < K) ? arow[ka]     : 0.f;
    a.y = (ka + 1 < K) ? arow[ka + 1] : 0.f;
    b.x = (ka     < K) ? W[(size_t)ka * N + nc]       : 0.f;
    b.y = (ka + 1 < K) ? W[(size_t)(ka + 1) * N + nc] : 0.f;
    acc = __builtin_amdgcn_wmma_f32_16x16x4_f32(false, a, false, b, (short)0, acc, false, false);
  }

  float bn = bias ? bias[nc] : 0.f;
#pragma unroll
  for (int r = 0; r < 8; ++r) {
    int row = m0 + r + 8 * half;
    float v = acc[r] + bn;
    if (act == 1) v = 0.5f * v * (1.f + erff(v * 0.70710678f));  // exact GELU
    C[(size_t)row * N + nc] = v;
  }
}

// ---------------- [rows,Kd] @ [Kd,1] + b -> sigmoid ----------------
__global__ void dot_sigmoid(const float* __restrict__ A, const float* __restrict__ w,
                            const float* __restrict__ b, float* __restrict__ out, int Kd) {
  int row = blockIdx.x, tid = threadIdx.x;
  __shared__ float red[256];
  float s = 0.f;
  for (int k = tid; k < Kd; k += 256) s += A[(size_t)row * Kd + k] * w[k];
  red[tid] = s; __syncthreads();
  for (int o = 128; o > 0; o >>= 1) { if (tid < o) red[tid] += red[tid + o]; __syncthreads(); }
  if (tid == 0) out[row] = 1.f / (1.f + expf(-(red[0] + b[0])));
}

__global__ void normalize_rows(float* __restrict__ X, int D) {
  int row = blockIdx.x, tid = threadIdx.x;
  __shared__ float red[256];
  float s = 0.f;
  for (int i = tid; i < D; i += 256) { float v = X[(size_t)row * D + i]; s += v * v; }
  red[tid] = s; __syncthreads();
  for (int o = 128; o > 0; o >>= 1) { if (tid < o) red[tid] += red[tid + o]; __syncthreads(); }
  float inv = 1.f / fmaxf(sqrtf(red[0]), 1e-6f);
  for (int i = tid; i < D; i += 256) X[(size_t)row * D + i] *= inv;
}

// ---------------- per-batch slot matching + feature assembly ----------------
__global__ void rung_match(const float* __restrict__ ck, int K, const float* __restrict__ cv, int V,
                           const float* __restrict__ context,
                           const float* __restrict__ keys, const float* __restrict__ values,
                           const float* __restrict__ conf, const float* __restrict__ age,
                           const float* __restrict__ alive,
                           const float* __restrict__ wp, const float* __restrict__ cp,
                           float* __restrict__ feat, int gd,
                           int* __restrict__ bestIdx, int* __restrict__ shIdx,
                           float* __restrict__ hasLiveF, int NS, float cadence) {
  int b = blockIdx.x, tid = threadIdx.x;
  __shared__ float score_s[8], alive_s[8], conf_s[8];
  __shared__ float s_best, s_cad, s_sur;
  __shared__ int s_bidx;
  int warp = tid >> 5, lane = tid & 31;
  if (warp < NS) {
    int s = warp;
    float al = alive[b * NS + s];
    float eps = (1.f - al) * 1e-4f;
    float ss = 0.f, dd = 0.f;
    for (int k = lane; k < K; k += 32) {
      float kv = keys[((size_t)b * NS + s) * K + k] + eps;
      ss += kv * kv; dd += kv * ck[b * K + k];
    }
    for (int off = 16; off > 0; off >>= 1) {
      ss += __shfl_xor(ss, off, 32);
      dd += __shfl_xor(dd, off, 32);
    }
    if (lane == 0) {
      score_s[s] = dd / fmaxf(sqrtf(ss), 1e-6f);
      alive_s[s] = al;
      conf_s[s] = conf[b * NS + s];
    }
  }
  __syncthreads();
  if (tid == 0) {
    bool hasl = false; float hl = 0.f;
    float bestv = -1e30f; int bi = 0;
    float occbest = 1e30f; int si = 0;
    for (int s = 0; s < NS; ++s) {
      bool live = alive_s[s] > 0.f;
      hasl = hasl || live;
      hl = fmaxf(hl, alive_s[s]);
      float sc = live ? score_s[s] : -1e4f;
      if (sc > bestv) { bestv = sc; bi = s; }      // first-occurrence argmax
      float occ = alive_s[s] * 1e4f + conf_s[s];
      if (occ < occbest) { occbest = occ; si = s; } // first-occurrence argmin
    }
    if (!hasl) { bi = 0; bestv = 0.f; }
    float mage = age[b * NS + bi];
    float cadp = 1.f / (1.f + expf(-(mage - cadence) / fmaxf(cadence, 1.f)));
    s_best = bestv; s_cad = cadp; s_sur = 1.f - bestv; s_bidx = bi;
    bestIdx[b] = bi; shIdx[b] = si; hasLiveF[b] = hl;
  }
  __syncthreads();
  int bi = s_bidx;
  float* f = feat + (size_t)b * gd;
  for (int i = tid; i < 256; i += 256) f[i] = context[b * 256 + i];
  for (int i = tid; i < K; i += 256) f[256 + i] = ck[b * K + i];
  for (int i = tid; i < V; i += 256) {
    f[256 + K + i] = cv[b * V + i];
    f[256 + K + V + i] = values[((size_t)b * NS + bi) * V + i];
  }
  if (tid == 0) {
    int o = 256 + K + 2 * V;
    f[o] = s_best; f[o + 1] = s_cad; f[o + 2] = s_sur; f[o + 3] = wp[b]; f[o + 4] = cp[b];
  }
}

// ---------------- per-batch state update + weighted summary ----------------
__global__ void rung_update(const float* __restrict__ cv, int V,
                            const float* __restrict__ values, const float* __restrict__ conf,
                            const float* __restrict__ alive,
                            const float* __restrict__ rg, const float* __restrict__ sg,
                            const float* __restrict__ tg,
                            const int* __restrict__ bestIdx, const int* __restrict__ shIdx,
                            const float* __restrict__ hasLiveF,
                            float* __restrict__ values_new, float* __restrict__ wscale,
                            float* __restrict__ summary, int NS,
                            float refresh_thr, float spawn_thr, float retire_thr, float conf_decay) {
  int b = blockIdx.x, tid = threadIdx.x;
  __shared__ float ww_s[8], wt_s[8];
  __shared__ float s_wsum;
  if (tid == 0) {
    float rraw = 1.f / (1.f + expf(-(rg[b] - refresh_thr) / TEMP)) * hasLiveF[b];
    float sraw = 1.f / (1.f + expf(-(sg[b] - spawn_thr) / TEMP));
    float denom = 1.f + rraw + sraw;
    float refresh = rraw / denom, spawn = sraw / denom;
    float retire = 1.f / (1.f + expf(-(tg[b] - retire_thr) / TEMP));
    int bi = bestIdx[b], si = shIdx[b];
    float wsum = 0.f;
    for (int s = 0; s < NS; ++s) {
      float rw = (s == bi) ? refresh : 0.f;
      float sw = (s == si) ? spawn : 0.f;
      float ww = fminf(rw + sw, 1.f);
      float tw = (s == bi) ? retire : 0.f;
      float cn = (conf[b * NS + s] * conf_decay * (1.f - ww) + ww) * (1.f - tw);
      float an = fminf(alive[b * NS + s] + sw, 1.f) * (1.f - tw);
      float wt = an * cn;
      ww_s[s] = ww; wt_s[s] = wt; wscale[b * NS + s] = wt; wsum += wt;
    }
    s_wsum = fmaxf(wsum, 1.f);
  }
  __syncthreads();
  for (int i = tid; i < NS * V; i += 256) {
    int s = i / V, v = i % V;
    float w = ww_s[s];
    values_new[((size_t)b * NS + s) * V + v] =
        values[((size_t)b * NS + s) * V + v] * (1.f - w) + cv[b * V + v] * w;
  }
  __syncthreads();
  for (int v = tid; v < V; v += 256) {
    float acc = 0.f;
    for (int s = 0; s < NS; ++s) acc += values_new[((size_t)b * NS + s) * V + v] * wt_s[s];
    summary[b * V + v] = acc / s_wsum;
  }
}

// ---------------- LayerNorm + scale, writing into [B,21,1024] out ----------------
__global__ void ln_out(const float* __restrict__ Y, const float* __restrict__ g,
                       const float* __restrict__ beta, const float* __restrict__ scale,
                       float* __restrict__ out, int NSr, int tokBase) {
  int row = blockIdx.x, tid = threadIdx.x;
  const float* y = Y + (size_t)row * 1024;
  __shared__ float red[256];
  float s = 0.f;
  for (int c = tid; c < 1024; c += 256) s += y[c];
  red[tid] = s; __syncthreads();
  for (int o = 128; o > 0; o >>= 1) { if (tid < o) red[tid] += red[tid + o]; __syncthreads(); }
  float mu = red[0] / 1024.f; __syncthreads();
  float s2 = 0.f;
  for (int c = tid; c < 1024; c += 256) { float d = y[c] - mu; s2 += d * d; }
  red[tid] = s2; __syncthreads();
  for (int o = 128; o > 0; o >>= 1) { if (tid < o) red[tid] += red[tid + o]; __syncthreads(); }
  float inv = rsqrtf(red[0] / 1024.f + 1e-5f);
  float sc = scale ? scale[row] : 1.f;
  int b = row / NSr, sl = row % NSr;
  float* op = out + ((size_t)(b * 21 + tokBase + sl)) * 1024;
  for (int c = tid; c < 1024; c += 256) op[c] = ((y[c] - mu) * inv * g[c] + beta[c]) * sc;
}

// ===================== host =====================
extern "C" void kernel_launch(void* const* d_in, const int* in_sizes, int n_in,
                              void* d_out, int out_size, void* d_ws, size_t ws_size,
                              hipStream_t stream) {
  (void)in_sizes; (void)n_in; (void)out_size; (void)ws_size;
  auto F = [&](int i) { return (const float*)d_in[i]; };
  const float* hidden = F(0);
  const int* amask = (const int*)d_in[1];
  float* out = (float*)d_out;

  float* ws = (float*)d_ws;
  size_t off = 0;
  auto alloc = [&](size_t n) { float* p = ws + off; off += (n + 63) & ~size_t(63); return p; };

  float* partial    = alloc((size_t)Bc * NCH * Hc);
  float* cnt        = alloc(Bc);
  float* pooledLast = alloc((size_t)Bc * 2 * Hc);
  float* hEnc       = alloc((size_t)Bc * 1024);
  float* context    = alloc((size_t)Bc * 256);
  float* lgh        = alloc((size_t)Bc * 384);
  float* write_p    = alloc(Bc);
  float* contra_p   = alloc(Bc);
  float* kh         = alloc((size_t)Bc * 512);
  float* ck         = alloc((size_t)Bc * 160);
  float* cv         = alloc((size_t)Bc * 320);
  float* feat       = alloc((size_t)Bc * 1061);
  float* gh         = alloc((size_t)Bc * 384);
  float* rg         = alloc(Bc);
  float* sg         = alloc(Bc);
  float* tg         = alloc(Bc);
  int*   bestIdx    = (int*)alloc(Bc);
  int*   shIdx      = (int*)alloc(Bc);
  float* hasLive    = alloc(Bc);
  float* values_new = alloc((size_t)Bc * 8 * 320);
  float* wscale     = alloc((size_t)Bc * 8);
  float* summary    = alloc((size_t)Bc * 320);
  float* sumY       = alloc((size_t)Bc * 1024);
  float* slotY      = alloc((size_t)Bc * 8 * 1024);

  auto gemm = [&](const float* A, const float* W, const float* bias, float* C,
                  int M, int N, int K, int act) {
    gemm16_wmma<<<dim3(N / 16, M / 16), 32, 0, stream>>>(A, W, bias, C, M, N, K, act);
  };

  // ---- pooling ----
  pool_partial<<<dim3(Hc / 256, NCH, Bc), 256, 0, stream>>>(hidden, amask, partial);
  mask_count<<<Bc, 256, 0, stream>>>(amask, cnt);
  pool_finalize<<<dim3(Hc / 256, Bc), 256, 0, stream>>>(partial, cnt, hidden, pooledLast);

  // ---- encoder: [32,2048] -> gelu 1024 -> 256 ----
  gemm(pooledLast, F(17), F(18), hEnc, Bc, 1024, 2048, 1);
  gemm(hEnc, F(19), F(20), context, Bc, 256, 1024, 0);

  // ---- ledger ----
  gemm(context, F(21), F(22), lgh, Bc, 384, 256, 1);
  dot_sigmoid<<<Bc, 256, 0, stream>>>(lgh, F(23), F(24), write_p, 384);
  gemm(context, F(25), F(26), lgh, Bc, 384, 256, 1);
  dot_sigmoid<<<Bc, 256, 0, stream>>>(lgh, F(27), F(28), contra_p, 384);

  // ---- rungs ----
  struct Spec { int NS, K, V, gd; float cadence; int st, pr, tok; };
  const Spec specs[3] = {
    {8,  96, 192, 256 +  96 + 2 * 192 + 5,  2.f,  2, 29,  0},
    {6, 128, 256, 256 + 128 + 2 * 256 + 5,  8.f,  7, 61,  9},
    {4, 160, 320, 256 + 160 + 2 * 320 + 5, 32.f, 12, 93, 16},
  };
  for (int r = 0; r < 3; ++r) {
    const Spec& sp = specs[r];
    const float* keys   = F(sp.st + 0);
    const float* values = F(sp.st + 1);
    const float* conf   = F(sp.st + 2);
    const float* age    = F(sp.st + 3);
    const float* alive  = F(sp.st + 4);
    int P = sp.pr;

    // key / value MLPs
    gemm(context, F(P + 0), F(P + 1), kh, Bc, 512, 256, 1);
    gemm(kh, F(P + 2), F(P + 3), ck, Bc, sp.K, 512, 0);
    normalize_rows<<<Bc, 256, 0, stream>>>(ck, sp.K);
    gemm(context, F(P + 4), F(P + 5), kh, Bc, 512, 256, 1);
    gemm(kh, F(P + 6), F(P + 7), cv, Bc, sp.V, 512, 0);

    // match + feature vector
    rung_match<<<Bc, 256, 0, stream>>>(ck, sp.K, cv, sp.V, context, keys, values, conf, age,
                                       alive, write_p, contra_p, feat, sp.gd,
                                       bestIdx, shIdx, hasLive, sp.NS, sp.cadence);

    // gates: refresh (+8), spawn (+12), retire (+20); promote is unused in output
    gemm(feat, F(P + 8), F(P + 9), gh, Bc, 384, sp.gd, 1);
    dot_sigmoid<<<Bc, 256, 0, stream>>>(gh, F(P + 10), F(P + 11), rg, 384);
    gemm(feat, F(P + 12), F(P + 13), gh, Bc, 384, sp.gd, 1);
    dot_sigmoid<<<Bc, 256, 0, stream>>>(gh, F(P + 14), F(P + 15), sg, 384);
    gemm(feat, F(P + 20), F(P + 21), gh, Bc, 384, sp.gd, 1);
    dot_sigmoid<<<Bc, 256, 0, stream>>>(gh, F(P + 22), F(P + 23), tg, 384);

    // state update
    rung_update<<<Bc, 256, 0, stream>>>(cv, sp.V, values, conf, alive, rg, sg, tg,
                                        bestIdx, shIdx, hasLive, values_new, wscale, summary,
                                        sp.NS, 0.55f, 0.6f, 0.15f, 0.995f);

    // summary token
    gemm(summary, F(P + 24), F(P + 25), sumY, Bc, 1024, sp.V, 0);
    ln_out<<<Bc, 256, 0, stream>>>(sumY, F(P + 26), F(P + 27), nullptr, out, 1, sp.tok);

    // slot tokens
    gemm(values_new, F(P + 28), F(P + 29), slotY, Bc * sp.NS, 1024, sp.V, 0);
    ln_out<<<Bc * sp.NS, 256, 0, stream>>>(slotY, F(P + 30), F(P + 31), wscale, out,
                                           sp.NS, sp.tok + 1);
  }
}